// GCN_27350351741210
// MI455X (gfx1250) — compile-verified
//
#include <hip/hip_runtime.h>
#include <math.h>

// ---------------------------------------------------------------------------
// GCN forward (2x GCNConv + ReLU + log_softmax) for MI455X / gfx1250.
// Dense GEMMs: V_WMMA_F32_16X16X4_F32, wave32, one wave = 16 x NPAD strip,
// W staged transposed+padded in LDS (no guards in the inner loop).
// ---------------------------------------------------------------------------

typedef __attribute__((ext_vector_type(2))) float v2f;
typedef __attribute__((ext_vector_type(8))) float v8f;

#define GCN_N_NODES 50000
#define GCN_N_EDGES 800000
#define GCN_IN_CH   128
#define GCN_HID_CH  64
#define GCN_OUT_CH  40

// ---------------- degree / normalization ----------------------------------

__global__ void gcn_zero_f32(float* __restrict__ p, int n) {
    int i = blockIdx.x * blockDim.x + threadIdx.x;
    if (i < n) p[i] = 0.0f;
}

__global__ void gcn_deg_count(const int* __restrict__ dst, float* __restrict__ deg) {
    int e = blockIdx.x * blockDim.x + threadIdx.x;
    if (e < GCN_N_EDGES) atomicAdd(&deg[dst[e]], 1.0f);
}

__global__ void gcn_deg_to_dis(float* __restrict__ deg) {
    int i = blockIdx.x * blockDim.x + threadIdx.x;
    if (i < GCN_N_NODES) deg[i] = rsqrtf(deg[i] + 1.0f);  // in-place -> deg_inv_sqrt
}

// ---------------- WMMA fp32 GEMM: H = A(MxK) @ W(KxNOUT) -------------------
// Block = WAVES waves; W is loaded once into LDS, transposed ([n][k]) and
// zero-padded to NPAD columns. Each wave computes one 16-row strip across all
// NPAD/16 N-tiles: one A fragment feeds NT independent WMMAs per k-step.

template <int K, int NOUT, int NPAD, int WAVES>
__global__ __launch_bounds__(32 * WAVES)
void gcn_gemm_f32(const float* __restrict__ A,
                  const float* __restrict__ W,
                  float* __restrict__ H,
                  int mtiles) {
    constexpr int NT = NPAD / 16;        // N tiles per wave
    constexpr int LDK = K + 2;           // pad: lanes land on distinct LDS banks
    __shared__ float WT[NPAD][LDK];      // W transposed: WT[n][k]

    // cooperative fill (coalesced reads of W, zero-pad n >= NOUT)
    for (int idx = threadIdx.x; idx < K * NPAD; idx += 32 * WAVES) {
        const int n = idx % NPAD;
        const int k = idx / NPAD;
        WT[n][k] = (n < NOUT) ? W[(size_t)k * NOUT + n] : 0.0f;
    }
    __syncthreads();

    const int wave = threadIdx.x >> 5;
    const int lane = threadIdx.x & 31;
    const int half = lane >> 4;          // 0: lanes 0-15, 1: lanes 16-31
    const int lid  = lane & 15;

    const int mtile = blockIdx.x * WAVES + wave;
    if (mtile >= mtiles) return;         // wave-uniform exit (EXEC stays all-1s)
    const int m0 = mtile * 16;

    const float* __restrict__ arow = A + (size_t)(m0 + lid) * K;

    v8f acc[NT];
#pragma unroll
    for (int t = 0; t < NT; ++t) acc[t] = (v8f){};

    for (int k0 = 0; k0 < K; k0 += 4) {
        const int kk = k0 + 2 * half;    // lanes 0-15: K=k0,k0+1 ; 16-31: k0+2,k0+3
        v2f a;
        a.x = arow[kk];
        a.y = arow[kk + 1];
#pragma unroll
        for (int t = 0; t < NT; ++t) {
            const v2f b = *reinterpret_cast<const v2f*>(&WT[t * 16 + lid][kk]);
            acc[t] = __builtin_amdgcn_wmma_f32_16x16x4_f32(false, a, false, b,
                                                           (short)0, acc[t],
                                                           false, false);
        }
    }

#pragma unroll
    for (int t = 0; t < NT; ++t) {
        const int col = t * 16 + lid;
        if (col < NOUT) {
#pragma unroll
            for (int v = 0; v < 8; ++v) {
                const int row = m0 + v + 8 * half;   // lanes 16-31 carry M+8
                H[(size_t)row * NOUT + col] = acc[t][v];
            }
        }
    }
}

// ---------------- aggregation: self-loop init + edge scatter ---------------

template <int C>
__global__ void gcn_self_loop_init(const float* __restrict__ H,
                                   const float* __restrict__ dis,
                                   float* __restrict__ agg) {
    unsigned i = blockIdx.x * blockDim.x + threadIdx.x;
    if (i < (unsigned)GCN_N_NODES * C) {
        const unsigned node = i / C;
        const float d = dis[node];
        agg[i] = H[i] * d * d;  // also initializes agg (overwrites poison)
    }
}

template <int C>
__global__ void gcn_edge_scatter(const int* __restrict__ src,
                                 const int* __restrict__ dst,
                                 const float* __restrict__ H,
                                 const float* __restrict__ dis,
                                 float* __restrict__ agg) {
    unsigned i = blockIdx.x * blockDim.x + threadIdx.x;
    if (i < (unsigned)GCN_N_EDGES * C) {
        const unsigned e = i / C;
        const unsigned c = i % C;
        const int s = src[e];
        const int d = dst[e];
        const float norm = dis[s] * dis[d];
        atomicAdd(&agg[(size_t)d * C + c], H[(size_t)s * C + c] * norm);
    }
}

// ---------------- epilogues ------------------------------------------------

__global__ void gcn_bias_relu(float* __restrict__ agg, const float* __restrict__ b) {
    unsigned i = blockIdx.x * blockDim.x + threadIdx.x;
    if (i < (unsigned)GCN_N_NODES * GCN_HID_CH) {
        const unsigned c = i & (GCN_HID_CH - 1);
        agg[i] = fmaxf(agg[i] + b[c], 0.0f);
    }
}

__global__ void gcn_bias_logsoftmax(const float* __restrict__ agg,
                                    const float* __restrict__ b,
                                    float* __restrict__ out) {
    int i = blockIdx.x * blockDim.x + threadIdx.x;
    if (i >= GCN_N_NODES) return;
    float v[GCN_OUT_CH];
    float m = -INFINITY;
#pragma unroll
    for (int c = 0; c < GCN_OUT_CH; ++c) {
        v[c] = agg[(size_t)i * GCN_OUT_CH + c] + b[c];
        m = fmaxf(m, v[c]);
    }
    float s = 0.0f;
#pragma unroll
    for (int c = 0; c < GCN_OUT_CH; ++c) s += expf(v[c] - m);
    const float ls = logf(s);
#pragma unroll
    for (int c = 0; c < GCN_OUT_CH; ++c)
        out[(size_t)i * GCN_OUT_CH + c] = v[c] - m - ls;
}

// ---------------- launcher -------------------------------------------------

extern "C" void kernel_launch(void* const* d_in, const int* in_sizes, int n_in,
                              void* d_out, int out_size, void* d_ws, size_t ws_size,
                              hipStream_t stream) {
    (void)in_sizes; (void)n_in; (void)out_size; (void)ws_size;

    const float* x  = (const float*)d_in[0];                 // [50000,128]
    const int*   ei = (const int*)d_in[1];                   // [2,800000] (int32)
    const float* W1 = (const float*)d_in[2];                 // [128,64]
    const float* b1 = (const float*)d_in[3];                 // [64]
    const float* W2 = (const float*)d_in[4];                 // [64,40]
    const float* b2 = (const float*)d_in[5];                 // [40]
    float*       out = (float*)d_out;                        // [50000,40]

    const int* src = ei;
    const int* dst = ei + GCN_N_EDGES;

    // workspace carve-out (floats)
    float* ws   = (float*)d_ws;
    float* dis  = ws;                                        // 50000 (deg -> rsqrt in place)
    float* H1   = ws + (1 << 16);                            // 50000*64
    float* AGG1 = H1 + (size_t)GCN_N_NODES * GCN_HID_CH;     // 50000*64
    float* H2   = AGG1 + (size_t)GCN_N_NODES * GCN_HID_CH;   // 50000*40
    float* AGG2 = H2 + (size_t)GCN_N_NODES * GCN_OUT_CH;     // 50000*40

    const int TB = 256;
    const int nodeBlocks = (GCN_N_NODES + TB - 1) / TB;
    const int edgeBlocks = (GCN_N_EDGES + TB - 1) / TB;
    const int MT = GCN_N_NODES / 16;                         // 3125 (exact)
    const int WAVES = 8;

    // degrees with self loops -> deg_inv_sqrt
    gcn_zero_f32<<<nodeBlocks, TB, 0, stream>>>(dis, GCN_N_NODES);
    gcn_deg_count<<<edgeBlocks, TB, 0, stream>>>(dst, dis);
    gcn_deg_to_dis<<<nodeBlocks, TB, 0, stream>>>(dis);

    // ---- layer 1: H1 = x @ W1 ; agg1 = D^-1/2 (A+I) D^-1/2 H1 ; relu(+b1)
    gcn_gemm_f32<GCN_IN_CH, GCN_HID_CH, 64, 8>
        <<<(MT + WAVES - 1) / WAVES, 32 * WAVES, 0, stream>>>(x, W1, H1, MT);

    {
        const unsigned n = (unsigned)GCN_N_NODES * GCN_HID_CH;
        gcn_self_loop_init<GCN_HID_CH><<<(n + TB - 1) / TB, TB, 0, stream>>>(H1, dis, AGG1);
        const unsigned m = (unsigned)GCN_N_EDGES * GCN_HID_CH;
        gcn_edge_scatter<GCN_HID_CH><<<(m + TB - 1) / TB, TB, 0, stream>>>(src, dst, H1, dis, AGG1);
        gcn_bias_relu<<<(n + TB - 1) / TB, TB, 0, stream>>>(AGG1, b1);
    }

    // ---- layer 2: H2 = relu(agg1) @ W2 ; agg2 ; log_softmax(+b2)
    gcn_gemm_f32<GCN_HID_CH, GCN_OUT_CH, 48, 8>
        <<<(MT + WAVES - 1) / WAVES, 32 * WAVES, 0, stream>>>(AGG1, W2, H2, MT);

    {
        const unsigned n = (unsigned)GCN_N_NODES * GCN_OUT_CH;
        gcn_self_loop_init<GCN_OUT_CH><<<(n + TB - 1) / TB, TB, 0, stream>>>(H2, dis, AGG2);
        const unsigned m = (unsigned)GCN_N_EDGES * GCN_OUT_CH;
        gcn_edge_scatter<GCN_OUT_CH><<<(m + TB - 1) / TB, TB, 0, stream>>>(src, dst, H2, dis, AGG2);
        gcn_bias_logsoftmax<<<nodeBlocks, TB, 0, stream>>>(AGG2, b2, out);
    }
}